// CustomModel_41961830481979
// MI455X (gfx1250) — compile-verified
//
#include <hip/hip_runtime.h>
#include <stdint.h>

typedef unsigned int u32;
typedef unsigned long long u64;
typedef __attribute__((ext_vector_type(4))) u32 v4u;
typedef __attribute__((ext_vector_type(4))) int  v4i;
typedef __attribute__((ext_vector_type(8))) int  v8i;

#define B_  1024
#define N_  8192
#define L_  3
#define E_  131072
#define TPB 1024   // 32 waves (wave32)

// ---------------------------------------------------------------------------
// TDM: 1-D async DMA of `nelem` fp32 from global -> LDS (ISA cdna5 §8 D# layout)
//   group0: count=1 | lds_addr | global_addr[56:0] | type=2
//   group1: data_size=2 (4B), tensor_dim0=nelem, tensor_dim1=1,
//           tile_dim0=nelem (<=65535), tile_dim1/2=0, dim0_stride=nelem
// Tracked with TENSORcnt; EXEC is ignored (wave-level op, uniform args).
// ---------------------------------------------------------------------------
__device__ __forceinline__ void tdm_load_1d(u32 lds_byte, const float* gptr, u32 nelem) {
  u64 ga = (u64)(uintptr_t)gptr;
  v4u g0;
  g0[0] = 1u;                                              // count=1, user mode
  g0[1] = lds_byte;                                        // lds_addr (bytes)
  g0[2] = (u32)ga;                                         // global_addr[31:0]
  g0[3] = ((u32)(ga >> 32) & 0x01FFFFFFu) | 0x80000000u;   // addr[56:32] | type=2
  v8i g1;
  g1[0] = 0x00020000;                                      // wg_mask=0, data_size=4B
  g1[1] = (int)((nelem & 0xFFFFu) << 16);                  // tensor_dim0[15:0]
  g1[2] = (int)(((nelem >> 16) & 0xFFFFu) | (1u << 16));   // tensor_dim0 hi, tensor_dim1=1
  g1[3] = (int)((nelem & 0xFFFFu) << 16);                  // tile_dim0 = nelem
  g1[4] = 0;                                               // tile_dim1=0, tile_dim2=0
  g1[5] = (int)nelem;                                      // tensor_dim0_stride lo
  g1[6] = 0;
  g1[7] = 0;
  v4i gz = {0, 0, 0, 0};
#if __clang_major__ >= 23
  v8i gz8 = {0, 0, 0, 0, 0, 0, 0, 0};
  __builtin_amdgcn_tensor_load_to_lds(g0, g1, gz, gz, gz8, 0);
#else
  __builtin_amdgcn_tensor_load_to_lds(g0, g1, gz, gz, 0);
#endif
}

__device__ __forceinline__ float apply_act(int a, float t) {
  switch (a) {
    case 0:  return t;
    case 1:  return t < 0.0f ? -1.0f : 1.0f;
    case 2:  return __sinf(t);
    case 3:  return __cosf(t);
    case 4:  return __expf(-t * t);
    case 5:  { float e = __expf(2.0f * t); return (e - 1.0f) / (e + 1.0f); } // tanh
    case 6:  return 1.0f / (1.0f + __expf(-t));                              // sigmoid
    case 7:  return fabsf(t);
    case 8:  return -t;
    default: return t > 0.0f ? t : 0.0f;                                     // relu
  }
}

// One block per batch row. LDS = (l+1)*N vals + N accumulator (<=128KB of 320KB/WGP).
__global__ void __launch_bounds__(TPB)
sparse_layer_kernel(const float* __restrict__ x,   const float* __restrict__ w,
                    const int*   __restrict__ src, const int*   __restrict__ dst,
                    const int*   __restrict__ act, float* __restrict__ ws,
                    float* __restrict__ out, int l)
{
  extern __shared__ float smem[];
  const int  b = blockIdx.x;
  const int  t = threadIdx.x;
  const u32  W = (u32)(l + 1) * N_;          // source-row width in floats
  float* vals = smem;                        // [0, W)
  float* z    = smem + W;                    // [W, W+N)

  const float* xrow  = x  + (size_t)b * N_;
  const float* wsrow = ws + (size_t)b * (2 * N_);   // layer-0/1 outputs live here

  // ---- 1) per-wave TDM async copy of the concatenated row into LDS --------
  {
    const u32 wid   = (u32)t >> 5;           // wave id, 0..31 (wave-uniform)
    const u32 chunk = W >> 5;                // W is a multiple of 32
    const u32 beg = wid * chunk, end = beg + chunk;
    const u32 lbase = (u32)(uintptr_t)(void*)smem;   // LDS byte offset
    if (beg < N_) {                          // part sourced from x
      u32 xe = end < N_ ? end : N_;
      tdm_load_1d(lbase + beg * 4u, xrow + beg, xe - beg);
    }
    if (end > N_) {                          // part sourced from ws (prev layers)
      u32 wb = beg > N_ ? beg : N_;
      tdm_load_1d(lbase + wb * 4u, wsrow + (wb - N_), end - wb);
    }
  }

  // ---- 2) zero the accumulator while the DMA is in flight -----------------
  for (u32 i = (u32)t; i < N_; i += TPB) z[i] = 0.0f;

  __builtin_amdgcn_s_wait_tensorcnt(0);      // wave's TDM writes landed in LDS
  __syncthreads();                           // make them visible block-wide

  // ---- 3) gather (LDS) -> conditional weight -> scatter-add (ds_add_f32) --
  {
    const int4*   s4 = (const int4*)  (src + (size_t)l * E_);
    const int4*   d4 = (const int4*)  (dst + (size_t)l * E_);
    const float4* w4 = (const float4*)(w   + (size_t)l * E_);
    const int nIter = E_ / 4;                // 32768 vec4 edges
    for (int i = t; i < nIter; i += TPB) {
      __builtin_prefetch(w4 + i + 4 * TPB, 0, 1);      // global_prefetch_b8 (L2-resident stream)
      int4   si = s4[i];
      int4   di = d4[i];
      float4 wi = w4[i];
      float v0 = vals[si.x], v1 = vals[si.y], v2 = vals[si.z], v3 = vals[si.w];
      float c0 = (v0 == 1.0f || v0 == -1.0f) ? v0 : v0 * wi.x;
      float c1 = (v1 == 1.0f || v1 == -1.0f) ? v1 : v1 * wi.y;
      float c2 = (v2 == 1.0f || v2 == -1.0f) ? v2 : v2 * wi.z;
      float c3 = (v3 == 1.0f || v3 == -1.0f) ? v3 : v3 * wi.w;
      atomicAdd(&z[di.x], c0);               // LDS float atomics, no-return form
      atomicAdd(&z[di.y], c1);
      atomicAdd(&z[di.z], c2);
      atomicAdd(&z[di.w], c3);
    }
  }
  __syncthreads();

  // ---- 4) per-column activation + writeback -------------------------------
  const int* arow = act + (size_t)l * N_;
  float* orow = (l == L_ - 1) ? (out + (size_t)b * N_)
                              : (ws + (size_t)b * (2 * N_) + (size_t)l * N_);
  for (int n = t; n < N_; n += TPB)
    orow[n] = apply_act(arow[n], z[n]);
}

extern "C" void kernel_launch(void* const* d_in, const int* in_sizes, int n_in,
                              void* d_out, int out_size, void* d_ws, size_t ws_size,
                              hipStream_t stream) {
  const float* x   = (const float*)d_in[0];   // (B, N)       fp32
  const float* w   = (const float*)d_in[1];   // (L, E)       fp32
  const int*   src = (const int*)  d_in[2];   // (L, E)       i32
  const int*   dst = (const int*)  d_in[3];   // (L, E)       i32
  const int*   act = (const int*)  d_in[4];   // (L, N)       i32
  float* out = (float*)d_out;                 // (B, N)       fp32
  float* ws  = (float*)d_ws;                  // needs B*2N*4 = 64 MB scratch

  for (int l = 0; l < L_; ++l) {
    size_t lds_bytes = (size_t)(l + 2) * N_ * sizeof(float);   // (l+1)*N vals + N accum
    hipLaunchKernelGGL(sparse_layer_kernel, dim3(B_), dim3(TPB), lds_bytes, stream,
                       x, w, src, dst, act, ws, out, l);
  }
}